// Model_58136677319042
// MI455X (gfx1250) — compile-verified
//
#include <hip/hip_runtime.h>
#include <stdint.h>

// ---------------------------------------------------------------------------
// MI455X (gfx1250) GCN forward: 4x [ (X @ W) -> edge scatter-add -> bias+PReLU ]
// HBM-bound (~400 MB @ 23.3 TB/s ~ 17us; 6.6 GFLOP of math), so full fp32 via
// V_WMMA_F32_16X16X4_F32. W (64 KB) is pre-permuted into WMMA fragment order,
// TDM-copied (tensor_load_to_lds) into LDS once per block, and consumed with
// two ds_load_b128 per K-step; the VMEM pipe then only streams X.
// ---------------------------------------------------------------------------

typedef float v2f  __attribute__((ext_vector_type(2)));
typedef float v8f  __attribute__((ext_vector_type(8)));
typedef unsigned int u32x4 __attribute__((ext_vector_type(4)));
typedef int   i32x8 __attribute__((ext_vector_type(8)));
typedef int   i32x4 __attribute__((ext_vector_type(4)));

#define N_NODES   50000
#define N_EDGES   800000
#define N_IN      256
#define N_H       64
#define ROW_TILES 3125                       // 50000 / 16
#define BLOCKS_PER_G ((ROW_TILES + 3) / 4)   // 782 blocks of 4 row-tiles
#define FTS_FLOATS ((size_t)4 * N_NODES * N_H)   // 12.8M f32 in d_ws
#define WSTAGE_FLOATS 16384                  // one 256x64 W in fragment order

// ---------------------------------------------------------------------------
// Kernel 0: permute W1,W2 into WMMA-fragment order so blocks can TDM-copy a
// contiguous 64 KB image into LDS.  staged[((kstep*32+lane)*4+t)*2+comp] =
// W[(4*kstep + 2*(lane>>4) + comp)*64 + t*16 + (lane&15)]   (a bijection).
// ---------------------------------------------------------------------------
__global__ __launch_bounds__(256) void stage_w(
    const float* __restrict__ w1, const float* __restrict__ w2,
    float* __restrict__ staged)
{
    const int i = blockIdx.x * 256 + threadIdx.x;        // 0 .. 32767
    const float* __restrict__ W = (i >> 14) ? w2 : w1;
    const int j     = i & 16383;
    const int comp  = j & 1;
    const int t     = (j >> 1) & 3;
    const int lane  = (j >> 3) & 31;
    const int kstep = j >> 8;
    const int k = 4 * kstep + 2 * (lane >> 4) + comp;
    const int n = t * 16 + (lane & 15);
    staged[i] = W[k * N_H + n];
}

// ---------------------------------------------------------------------------
// TDM: contiguous 1-D copy of `nelem` f32 from global to LDS (ISA 08 §8).
// Group0: count=1 | lds_addr | global_addr[56:0] | type=2.
// Group1: data_size=4B, tensor_dim0=tile_dim0=nelem, tensor_dim1=tile_dim1=1.
// Groups 2/3 zero (<=2-D tensor).  Tracked by TENSORcnt.
// This toolchain exposes the 6-arg builtin: (g0, g1, g2, g3, i32x8, cpol).
// ---------------------------------------------------------------------------
static __device__ __forceinline__ void tdm_copy_to_lds(uint32_t lds_addr,
                                                       uint64_t gaddr,
                                                       uint32_t nelem)
{
    u32x4 g0;
    g0[0] = 1u;                                              // count=1, user D#
    g0[1] = lds_addr;                                        // lds_addr[31:0]
    g0[2] = (uint32_t)(gaddr & 0xFFFFFFFFu);                 // global_addr[31:0]
    g0[3] = (uint32_t)((gaddr >> 32) & 0x01FFFFFFu)          // global_addr[56:32]
          | (2u << 30);                                      // type = 2 ("image")
    i32x8 g1;
    g1[0] = (int)(2u << 16);                                 // data_size = 4 B
    g1[1] = (int)((nelem & 0xFFFFu) << 16);                  // tensor_dim0[15:0]
    g1[2] = (int)(((nelem >> 16) & 0xFFFFu) | (1u << 16));   // dim0[31:16] | tensor_dim1=1
    g1[3] = (int)((nelem & 0xFFFFu) << 16);                  // tile_dim0
    g1[4] = 1;                                               // tile_dim1 = 1
    g1[5] = (int)nelem;                                      // tensor_dim0_stride[31:0]
    g1[6] = 0;
    g1[7] = 0;
    i32x4 z4 = {0, 0, 0, 0};
    i32x8 z8 = {0, 0, 0, 0, 0, 0, 0, 0};
    __builtin_amdgcn_tensor_load_to_lds(g0, g1, z4, z4, z8, 0);
}

// ---------------------------------------------------------------------------
// Kernel 1: fts[g] = X[g] @ W[g]  (50000x256 @ 256x64), fp32 WMMA.
// Block = 4 waves, each owning a 16-row x 64-col strip (4 accumulators share
// one A fragment -> 4x A reuse).  Each wave TDM-copies a 16 KB quadrant of
// the fragment-ordered W into LDS; inner loop = 1 global b64 (A) +
// 2 ds_load_b128 (all four B fragments) + 4 WMMA per K-step.
// ---------------------------------------------------------------------------
__global__ __launch_bounds__(128) void gcn_gemm_wmma(
    const float* __restrict__ x0, const float* __restrict__ x1,
    const float* __restrict__ x2, const float* __restrict__ x3,
    const float* __restrict__ stagedW,   // [2][16384] fragment-ordered W1,W2
    float* __restrict__ fts)
{
    __shared__ __align__(16) float sW[WSTAGE_FLOATS];        // 64 KB

    const int lane = threadIdx.x & 31;
    const int wv   = threadIdx.x >> 5;                       // wave in block
    const int g    = blockIdx.x / BLOCKS_PER_G;              // graph 0..3
    const int rt   = (blockIdx.x % BLOCKS_PER_G) * 4 + wv;   // 16-row tile

    // --- TDM stage: each wave copies its 16 KB quadrant of W into LDS ---
    {
        const uint64_t gsrc =
            (uint64_t)(uintptr_t)(stagedW + (size_t)(g & 1) * WSTAGE_FLOATS + wv * 4096);
        const uint32_t ldst = (uint32_t)(uintptr_t)sW + (uint32_t)wv * 16384u;
        tdm_copy_to_lds(ldst, gsrc, 4096);
        __builtin_amdgcn_s_wait_tensorcnt(0);
    }
    __syncthreads();

    if (rt < ROW_TILES) {                                    // wave-uniform guard
        const float* __restrict__ X = (g == 0) ? x0 : (g == 1) ? x1 : (g == 2) ? x2 : x3;

        const int lo = lane & 15;        // M (A) / N (B, C/D)
        const int hi = lane >> 4;        // selects K pair {k0,k0+1} vs {k0+2,k0+3}
        const float* __restrict__ arow = X + (size_t)(rt * 16 + lo) * N_IN;
        const float4* __restrict__ sW4 = (const float4*)sW;

        v8f acc0 = {}, acc1 = {}, acc2 = {}, acc3 = {};

        for (int kstep = 0; kstep < N_IN / 4; ++kstep) {
            // A fragment: two consecutive K values, one 8-byte global load.
            v2f a = *(const v2f*)(arow + kstep * 4 + 2 * hi);

            // All four B fragments in 32 contiguous LDS bytes per lane.
            const int fb = (kstep * 32 + lane) * 2;          // float4 index
            float4 q0 = sW4[fb + 0];                         // t0, t1
            float4 q1 = sW4[fb + 1];                         // t2, t3
            v2f b0 = {q0.x, q0.y};
            v2f b1 = {q0.z, q0.w};
            v2f b2 = {q1.x, q1.y};
            v2f b3 = {q1.z, q1.w};

            acc0 = __builtin_amdgcn_wmma_f32_16x16x4_f32(false, a, false, b0, (short)0, acc0, false, false);
            acc1 = __builtin_amdgcn_wmma_f32_16x16x4_f32(false, a, false, b1, (short)0, acc1, false, false);
            acc2 = __builtin_amdgcn_wmma_f32_16x16x4_f32(false, a, false, b2, (short)0, acc2, false, false);
            acc3 = __builtin_amdgcn_wmma_f32_16x16x4_f32(false, a, false, b3, (short)0, acc3, false, false);
        }

        // D layout: VGPR r -> row (rt*16 + hi*8 + r), col t*16 + lo.
        float* __restrict__ out = fts + ((size_t)g * N_NODES + rt * 16 + hi * 8) * N_H;
        #pragma unroll
        for (int r = 0; r < 8; ++r) {
            float* __restrict__ orow = out + (size_t)r * N_H;
            orow[ 0 + lo] = acc0[r];
            orow[16 + lo] = acc1[r];
            orow[32 + lo] = acc2[r];
            orow[48 + lo] = acc3[r];
        }
    }
}

// ---------------------------------------------------------------------------
// Kernel 2: edge scatter-add.  64 threads per edge, one hidden channel each:
// coalesced 256 B gather of fts[col], global f32 atomic add into out[row].
// ---------------------------------------------------------------------------
__global__ __launch_bounds__(256) void gcn_scatter(
    const int*   __restrict__ adj_rows,  const int*   __restrict__ adj_cols,
    const float* __restrict__ adj_w,
    const int*   __restrict__ diff_rows, const int*   __restrict__ diff_cols,
    const float* __restrict__ diff_w,
    const float* __restrict__ fts, float* __restrict__ out)
{
    const int slot = blockIdx.x * 4 + (threadIdx.x >> 6);   // 0 .. 4*N_EDGES-1
    const int h    = threadIdx.x & 63;
    const int g    = slot / N_EDGES;
    const int e    = slot % N_EDGES;

    const int*   __restrict__ rows = (g & 1) ? diff_rows : adj_rows;
    const int*   __restrict__ cols = (g & 1) ? diff_cols : adj_cols;
    const float* __restrict__ w    = (g & 1) ? diff_w    : adj_w;

    const int   r  = rows[e];
    const int   c  = cols[e];
    const float wv = w[e];

    const float val = wv * fts[((size_t)g * N_NODES + c) * N_H + h];
    atomicAdd(out + ((size_t)g * N_NODES + r) * N_H + h, val);
}

// ---------------------------------------------------------------------------
// Kernel 3: out = PReLU(out + b).  float4-vectorized epilogue.
// ---------------------------------------------------------------------------
__global__ __launch_bounds__(256) void gcn_bias_prelu(
    const float* __restrict__ b1, const float* __restrict__ a1,
    const float* __restrict__ b2, const float* __restrict__ a2,
    float* __restrict__ out)
{
    const size_t j = ((size_t)blockIdx.x * blockDim.x + threadIdx.x) * 4;
    const int g = (int)(j / ((size_t)N_NODES * N_H));
    const int h = (int)(j % N_H);

    const float* __restrict__ b = (g & 1) ? b2 : b1;
    const float  a              = (g & 1) ? *a2 : *a1;

    float4 v = *(float4*)(out + j);
    v.x += b[h + 0];  v.y += b[h + 1];  v.z += b[h + 2];  v.w += b[h + 3];
    v.x = (v.x >= 0.f) ? v.x : a * v.x;
    v.y = (v.y >= 0.f) ? v.y : a * v.y;
    v.z = (v.z >= 0.f) ? v.z : a * v.z;
    v.w = (v.w >= 0.f) ? v.w : a * v.w;
    *(float4*)(out + j) = v;
}

// ---------------------------------------------------------------------------
extern "C" void kernel_launch(void* const* d_in, const int* in_sizes, int n_in,
                              void* d_out, int out_size, void* d_ws, size_t ws_size,
                              hipStream_t stream)
{
    (void)in_sizes; (void)n_in; (void)ws_size;

    const float* x0        = (const float*)d_in[0];   // bf
    const float* x1        = (const float*)d_in[1];   // bl
    const float* x2        = (const float*)d_in[2];   // shuf_fts
    const float* x3        = (const float*)d_in[3];   // shuf_fls
    const int*   adj_rows  = (const int*)  d_in[4];
    const int*   adj_cols  = (const int*)  d_in[5];
    const float* adj_w     = (const float*)d_in[6];
    const int*   diff_rows = (const int*)  d_in[7];
    const int*   diff_cols = (const int*)  d_in[8];
    const float* diff_w    = (const float*)d_in[9];
    const float* W1        = (const float*)d_in[10];
    const float* b1        = (const float*)d_in[11];
    const float* a1        = (const float*)d_in[12];
    const float* W2        = (const float*)d_in[13];
    const float* b2        = (const float*)d_in[14];
    const float* a2        = (const float*)d_in[15];

    float* out     = (float*)d_out;                   // [4][1][N_NODES][N_H]
    float* fts     = (float*)d_ws;                    // [4][N_NODES][N_H]
    float* stagedW = (float*)d_ws + FTS_FLOATS;       // [2][16384]

    // Aggregation target must start at zero every call (deterministic).
    (void)hipMemsetAsync(d_out, 0, (size_t)out_size * sizeof(float), stream);

    // 0) Permute W1,W2 into WMMA fragment order (32768 elements).
    stage_w<<<128, 256, 0, stream>>>(W1, W2, stagedW);

    // 1) Four GEMMs via fp32 WMMA with TDM-staged B in LDS.
    gcn_gemm_wmma<<<4 * BLOCKS_PER_G, 128, 0, stream>>>(x0, x1, x2, x3, stagedW, fts);

    // 2) Edge scatter: 4 graphs * 800K edges, 4 edges / 256-thread block.
    gcn_scatter<<<N_EDGES, 256, 0, stream>>>(adj_rows, adj_cols, adj_w,
                                             diff_rows, diff_cols, diff_w,
                                             fts, out);

    // 3) Bias + PReLU over 12.8M outputs, float4 per thread.
    gcn_bias_prelu<<<(4 * N_NODES * N_H / 4) / 256, 256, 0, stream>>>(b1, a1, b2, a2, out);
}